// ContrastiveTreeLoss_27238682591477
// MI455X (gfx1250) — compile-verified
//
#include <hip/hip_runtime.h>
#include <hip/hip_bf16.h>

// Problem constants (match the reference's setup_inputs()).
constexpr int BB = 512;   // batch (sentences)
constexpr int NN = 256;   // tokens per sentence
constexpr int KK = 4;     // negatives per sentence
constexpr float MARGIN_F = 2.0f;

typedef __attribute__((ext_vector_type(2))) float v2f;
typedef __attribute__((ext_vector_type(8))) float v8f;

// Exact f32 wave32 all-reduce on the CDNA5 matrix pipe.
// A = [x | 0] per lane, B = ones(4x16)  =>  D[m,*] = x(lane m) + x(lane m+16).
// D VGPR v holds S_v in lanes 0-15 and S_{v+8} in lanes 16-31 (16x16 f32 C/D
// layout), so summing the 8 accumulator VGPRs leaves the two half-wave sums in
// the two lane halves; one xor-16 shuffle completes a broadcast reduction.
__device__ __forceinline__ float wave_allreduce_f32(float x) {
    v2f a;  a[0] = x;    a[1] = 0.0f;
    v2f bm; bm[0] = 1.0f; bm[1] = 1.0f;
    v8f c = {};
    c = __builtin_amdgcn_wmma_f32_16x16x4_f32(
            /*neg_a=*/false, a, /*neg_b=*/false, bm,
            /*c_mod=*/(short)0, c, /*reuse_a=*/false, /*reuse_b=*/false);
    float t = c[0] + c[1] + c[2] + c[3] + c[4] + c[5] + c[6] + c[7];
    t += __shfl_xor(t, 16, 32);
    return t;  // broadcast to all 32 lanes
}

__device__ __forceinline__ int clamp_head(int h) {
    return h < 0 ? 0 : (h > NN - 1 ? NN - 1 : h);
}

// One block per sentence b. Thread d owns dependent column d (d=0 excluded).
// Each thread accumulates r_k = m * (neg_val_k - gold_val); block-reduces the
// four r_k and writes loss_b = sum_k max(MARGIN + R_k, 0) to ws[b].
__global__ __launch_bounds__(256) void tree_margin_kernel(
        const float* __restrict__ arc,   // [BB, NN, NN]
        const int*   __restrict__ gold,  // [BB, NN]
        const int*   __restrict__ mask,  // [BB, NN]
        const int*   __restrict__ neg,   // [KK, BB, NN]
        float*       __restrict__ ws)    // [BB]
{
    const int b = blockIdx.x;
    const int d = threadIdx.x;                 // 0..255
    const float* __restrict__ arcb = arc + (size_t)b * (NN * NN);

    float r[KK] = {0.0f, 0.0f, 0.0f, 0.0f};
    if (d >= 1) {
        const float m  = (float)mask[b * NN + d];
        const int   gh = clamp_head(gold[b * NN + d]);
        const float gv = arcb[(size_t)gh * NN + d] * m;   // masked gold score
#pragma unroll
        for (int k = 0; k < KK; ++k) {
            const int h = clamp_head(neg[((size_t)k * BB + b) * NN + d]);
            r[k] = arcb[(size_t)h * NN + d] * m - gv;
        }
    }

    __shared__ float lds[KK * 256];
#pragma unroll
    for (int k = 0; k < KK; ++k) lds[k * 256 + d] = r[k];
    __syncthreads();

    // All 8 waves redundantly reduce the same LDS tile: identical results,
    // no divergence, EXEC is all-ones at every WMMA (ISA requirement).
    const int lane = d & 31;
    float loss = 0.0f;
#pragma unroll
    for (int k = 0; k < KK; ++k) {
        float x = 0.0f;
#pragma unroll
        for (int j = 0; j < 8; ++j) x += lds[k * 256 + lane + 32 * j];
        const float Rk = wave_allreduce_f32(x);   // = neg_total_k - gold_total
        loss += fmaxf(MARGIN_F + Rk, 0.0f);
    }
    if (d == 0) ws[b] = loss;
}

// Deterministic single-block tree reduction of the 512 per-sentence losses.
__global__ __launch_bounds__(256) void final_reduce_kernel(
        const float* __restrict__ ws, float* __restrict__ out)
{
    const int t = threadIdx.x;
    __shared__ float lds[256];
    lds[t] = ws[t] + ws[t + 256];
    __syncthreads();

    const int lane = t & 31;
    float x = 0.0f;
#pragma unroll
    for (int j = 0; j < 8; ++j) x += lds[lane + 32 * j];
    const float total = wave_allreduce_f32(x);
    if (t == 0) out[0] = total * (1.0f / (float)(KK * BB));  // mean over (K,B)
}

extern "C" void kernel_launch(void* const* d_in, const int* in_sizes, int n_in,
                              void* d_out, int out_size, void* d_ws, size_t ws_size,
                              hipStream_t stream) {
    (void)in_sizes; (void)n_in; (void)out_size; (void)ws_size;
    const float* arc  = (const float*)d_in[0];  // arc_scores [512,256,256] f32
    const int*   gold = (const int*)d_in[1];    // gold_heads [512,256] i32
    const int*   mask = (const int*)d_in[2];    // mask       [512,256] i32
    const int*   neg  = (const int*)d_in[3];    // neg_heads  [4,512,256] i32
    float* ws  = (float*)d_ws;                  // 512 floats of scratch
    float* out = (float*)d_out;                 // scalar f32

    tree_margin_kernel<<<BB, 256, 0, stream>>>(arc, gold, mask, neg, ws);
    final_reduce_kernel<<<1, 256, 0, stream>>>(ws, out);
}